// Grid_6382321401985
// MI455X (gfx1250) — compile-verified
//
#include <hip/hip_runtime.h>
#include <stdint.h>

#define NN 2304
#define N2 (NN * NN)

typedef __attribute__((ext_vector_type(2))) float v2f;
typedef __attribute__((ext_vector_type(8))) float v8f;

// Monotone float -> uint key (works for our domain: values >= -2)
__device__ __forceinline__ unsigned int fkey(float v) {
    unsigned int u = __float_as_uint(v);
    return (u & 0x80000000u) ? ~u : (u | 0x80000000u);
}

// ---------------------------------------------------------------------------
// GEMM via V_WMMA_F32_16X16X4_F32.  MODE 0: T = A*B stored.  MODE 1: compute
// A*B tile and dot with Tin, emit per-block partial sum (deterministic).
// ---------------------------------------------------------------------------
template <int MODE>
__global__ __launch_bounds__(256) void k_gemm(const float* __restrict__ A,
                                              const float* __restrict__ B,
                                              float* __restrict__ Tout,
                                              const float* __restrict__ Tin,
                                              float* __restrict__ partial) {
    __shared__ __align__(16) float As[128 * 34];  // [row][k], stride 34 (even -> 8B aligned pairs)
    __shared__ __align__(16) float Bs[128 * 34];  // transposed: [col][k]
    __shared__ float red[256];

    const int tid  = threadIdx.x;
    const int wv   = tid >> 5;
    const int lane = tid & 31;
    const int hi   = lane >> 4;  // half-wave select (K pair)
    const int ln   = lane & 15;
    const int rowBase = blockIdx.y * 128;
    const int colBase = blockIdx.x * 128;

    v8f acc[8];
#pragma unroll
    for (int s = 0; s < 8; s++)
#pragma unroll
        for (int v = 0; v < 8; v++) acc[s][v] = 0.0f;

    for (int k0 = 0; k0 < NN; k0 += 32) {
#pragma unroll
        for (int i = 0; i < 16; i++) {  // A tile 128x32
            int flat = i * 256 + tid;
            int r = flat >> 5, c = flat & 31;
            As[r * 34 + c] = A[(size_t)(rowBase + r) * NN + k0 + c];
        }
#pragma unroll
        for (int i = 0; i < 16; i++) {  // B tile 32x128, stored transposed
            int flat = i * 256 + tid;
            int kk = flat >> 7, n = flat & 127;
            Bs[n * 34 + kk] = B[(size_t)(k0 + kk) * NN + colBase + n];
        }
        __syncthreads();
#pragma unroll
        for (int k = 0; k < 32; k += 4) {
            // A frag: lane(hi,ln) -> row = ln, elems K=k+2*hi, k+2*hi+1
            v2f a = *(const v2f*)&As[(wv * 16 + ln) * 34 + k + 2 * hi];
#pragma unroll
            for (int s = 0; s < 8; s++) {
                // B frag: lane(hi,ln) -> col = ln, elems K=k+2*hi, k+2*hi+1
                v2f b = *(const v2f*)&Bs[(s * 16 + ln) * 34 + k + 2 * hi];
                acc[s] = __builtin_amdgcn_wmma_f32_16x16x4_f32(
                    false, a, false, b, (short)0, acc[s], false, false);
            }
        }
        __syncthreads();
    }

    if (MODE == 0) {
#pragma unroll
        for (int s = 0; s < 8; s++)
#pragma unroll
            for (int v = 0; v < 8; v++)
                Tout[(size_t)(rowBase + wv * 16 + v + 8 * hi) * NN + colBase + s * 16 + ln] =
                    acc[s][v];
    } else {
        float loc = 0.0f;
#pragma unroll
        for (int s = 0; s < 8; s++)
#pragma unroll
            for (int v = 0; v < 8; v++)
                loc += acc[s][v] *
                       Tin[(size_t)(rowBase + wv * 16 + v + 8 * hi) * NN + colBase + s * 16 + ln];
        red[tid] = loc;
        __syncthreads();
        for (int off = 128; off > 0; off >>= 1) {
            if (tid < off) red[tid] += red[tid + off];
            __syncthreads();
        }
        if (tid == 0) partial[blockIdx.y * gridDim.x + blockIdx.x] = red[0];
    }
}

// ---------------------------------------------------------------------------
// ras() pieces
// ---------------------------------------------------------------------------
__global__ __launch_bounds__(256) void k_rowsum(const float* __restrict__ g,
                                                float* __restrict__ rowsum) {
    __shared__ float red[256];
    int r = blockIdx.x;
    float s = 0.0f;
    for (int c = threadIdx.x; c < NN; c += 256) s += g[(size_t)r * NN + c];
    red[threadIdx.x] = s;
    __syncthreads();
    for (int off = 128; off > 0; off >>= 1) {
        if (threadIdx.x < off) red[threadIdx.x] += red[threadIdx.x + off];
        __syncthreads();
    }
    if (threadIdx.x == 0) rowsum[r] = red[0];
}

__global__ __launch_bounds__(256) void k_colsum(const float* __restrict__ g,
                                                const float* __restrict__ rowsum,
                                                float* __restrict__ colsum) {
    int d = blockIdx.x * 256 + threadIdx.x;
    if (d >= NN) return;
    float s = 0.0f;
    for (int a = 0; a < NN; a++) s += g[(size_t)a * NN + d] / rowsum[a];
    colsum[d] = s;
}

__global__ __launch_bounds__(256) void k_ras(const float* __restrict__ g,
                                             const float* __restrict__ rowsum,
                                             const float* __restrict__ colsum,
                                             float* __restrict__ partial) {
    __shared__ float red[256];
    float acc = 0.0f;
    int stride = gridDim.x * 256;
    for (int idx = blockIdx.x * 256 + threadIdx.x; idx < N2; idx += stride) {
        int a = idx / NN, d = idx - a * NN;
        float gv  = g[idx];
        float ras = (gv / rowsum[a]) / colsum[d];
        float df  = gv - ras;
        acc += df * df;
    }
    red[threadIdx.x] = acc;
    __syncthreads();
    for (int off = 128; off > 0; off >>= 1) {
        if (threadIdx.x < off) red[threadIdx.x] += red[threadIdx.x + off];
        __syncthreads();
    }
    if (threadIdx.x == 0) partial[blockIdx.x] = red[0];
}

// ---------------------------------------------------------------------------
// Greedy discretization
// ---------------------------------------------------------------------------
__global__ __launch_bounds__(256) void k_rowmax(const float* __restrict__ g,
                                                unsigned int* __restrict__ rowKey,
                                                int* __restrict__ rowArg) {
    __shared__ unsigned long long best;
    int r = blockIdx.x;
    if (threadIdx.x == 0) best = 0ull;
    __syncthreads();
    unsigned long long loc = 0ull;
    for (int c = threadIdx.x; c < NN; c += 256) {
        unsigned int k = fkey(fabsf(g[(size_t)r * NN + c]));
        unsigned long long cand = ((unsigned long long)k << 32) | (unsigned int)(~c);
        if (cand > loc) loc = cand;
    }
    atomicMax(&best, loc);
    __syncthreads();
    if (threadIdx.x == 0) {
        rowKey[r] = (unsigned int)(best >> 32);
        rowArg[r] = (int)(~(unsigned int)best);
    }
}

__global__ __launch_bounds__(1024) void k_greedy(const float* __restrict__ g,
                                                 const unsigned int* __restrict__ rowKeyIn,
                                                 const int* __restrict__ rowArgIn,
                                                 int* __restrict__ perm) {
    __shared__ unsigned int rowKey[NN];
    __shared__ int rowArg[NN];
    __shared__ int colAct[NN];
    __shared__ int affList[NN];
    __shared__ int affCount;
    __shared__ unsigned long long best;
    __shared__ int curJ;

    const int tid  = threadIdx.x;
    const int lane = tid & 31;
    const int wv   = tid >> 5;

    for (int r = tid; r < NN; r += 1024) {
        rowKey[r] = rowKeyIn[r];
        rowArg[r] = rowArgIn[r];
        colAct[r] = 1;
    }
    __syncthreads();

    for (int t = 0; t < NN; t++) {
        if (tid == 0) { best = 0ull; affCount = 0; }
        __syncthreads();
        // global argmax over active row maxima (val desc, row asc)
        unsigned long long loc = 0ull;
        for (int r = tid; r < NN; r += 1024) {
            unsigned int k = rowKey[r];
            if (k) {
                unsigned long long cand = ((unsigned long long)k << 32) | (unsigned int)(~r);
                if (cand > loc) loc = cand;
            }
        }
        atomicMax(&best, loc);
        __syncthreads();
        if (tid == 0) {
            int i = (int)(~(unsigned int)best);
            int j = rowArg[i];
            perm[i]   = j;
            rowKey[i] = 0;   // eliminate row
            colAct[j] = 0;   // eliminate column
            curJ      = j;
        }
        __syncthreads();
        int j = curJ;
        // rows whose cached argmax pointed at the removed column
        for (int r = tid; r < NN; r += 1024) {
            if (rowKey[r] && rowArg[r] == j) {
                int p = atomicAdd(&affCount, 1);
                affList[p] = r;
            }
        }
        __syncthreads();
        int cnt = affCount;
        for (int idx = wv; idx < cnt; idx += 32) {  // one wave per affected row
            int r = affList[idx];
            unsigned long long lb = 0ull;
            for (int c = lane; c < NN; c += 32) {
                if (colAct[c]) {
                    unsigned int k = fkey(fabsf(g[(size_t)r * NN + c]));
                    unsigned long long cand = ((unsigned long long)k << 32) | (unsigned int)(~c);
                    if (cand > lb) lb = cand;
                }
            }
            for (int off = 16; off > 0; off >>= 1) {
                unsigned long long o = __shfl_down(lb, off, 32);
                if (o > lb) lb = o;
            }
            if (lane == 0) {
                rowKey[r] = (unsigned int)(lb >> 32);
                rowArg[r] = (int)(~(unsigned int)lb);
            }
        }
        __syncthreads();
    }
}

__global__ __launch_bounds__(256) void k_discrete(const int* __restrict__ perm,
                                                  float* __restrict__ out) {
    int stride = gridDim.x * 256;
    for (int idx = blockIdx.x * 256 + threadIdx.x; idx < N2; idx += stride) {
        int a = idx / NN, c = idx - a * NN;
        out[idx] = (c == perm[a]) ? 1.0f : 0.0f;
    }
}

// real_loss = sum_{a,b} dist[p_a, p_b] * dist[a, b]
__global__ __launch_bounds__(256) void k_realloss(const float* __restrict__ dist,
                                                  const int* __restrict__ perm,
                                                  float* __restrict__ partial) {
    __shared__ float red[256];
    float acc = 0.0f;
    int stride = gridDim.x * 256;
    for (int idx = blockIdx.x * 256 + threadIdx.x; idx < N2; idx += stride) {
        int a = idx / NN, b = idx - a * NN;
        acc += dist[(size_t)perm[a] * NN + perm[b]] * dist[idx];
    }
    red[threadIdx.x] = acc;
    __syncthreads();
    for (int off = 128; off > 0; off >>= 1) {
        if (threadIdx.x < off) red[threadIdx.x] += red[threadIdx.x + off];
        __syncthreads();
    }
    if (threadIdx.x == 0) partial[blockIdx.x] = red[0];
}

__global__ __launch_bounds__(256) void k_final(const float* __restrict__ pGL, int nGL,
                                               const float* __restrict__ pRAS, int nRAS,
                                               const float* __restrict__ pRL, int nRL,
                                               const float* __restrict__ ratioPtr,
                                               float* __restrict__ out) {
    __shared__ float red[256];
    int tid = threadIdx.x;
    float a;

    a = 0.0f;
    for (int i = tid; i < nGL; i += 256) a += pGL[i];
    red[tid] = a; __syncthreads();
    for (int off = 128; off > 0; off >>= 1) { if (tid < off) red[tid] += red[tid + off]; __syncthreads(); }
    float gl = red[0]; __syncthreads();

    a = 0.0f;
    for (int i = tid; i < nRAS; i += 256) a += pRAS[i];
    red[tid] = a; __syncthreads();
    for (int off = 128; off > 0; off >>= 1) { if (tid < off) red[tid] += red[tid + off]; __syncthreads(); }
    float ras = red[0]; __syncthreads();

    a = 0.0f;
    for (int i = tid; i < nRL; i += 256) a += pRL[i];
    red[tid] = a; __syncthreads();
    for (int off = 128; off > 0; off >>= 1) { if (tid < off) red[tid] += red[tid + off]; __syncthreads(); }
    float rl = red[0];

    if (tid == 0) {
        float ratio = ratioPtr[0];
        out[0] = gl * (1.0f - ratio) + ras * ratio;
        out[1] = rl;
    }
}

// ---------------------------------------------------------------------------
extern "C" void kernel_launch(void* const* d_in, const int* in_sizes, int n_in,
                              void* d_out, int out_size, void* d_ws, size_t ws_size,
                              hipStream_t stream) {
    const float* grid  = (const float*)d_in[0];
    const float* dist  = (const float*)d_in[1];
    const float* ratio = (const float*)d_in[2];
    float* out = (float*)d_out;

    // workspace layout (~21.3 MB)
    float* T             = (float*)d_ws;        // N2
    float* rowsum        = T + N2;              // NN
    float* colsum        = rowsum + NN;         // NN
    unsigned int* rowKey = (unsigned int*)(colsum + NN);  // NN
    int* rowArg          = (int*)(rowKey + NN);           // NN
    int* perm            = rowArg + NN;                   // NN
    float* pGL           = (float*)(perm + NN); // 324
    float* pRAS          = pGL + 324;           // 1024
    float* pRL           = pRAS + 1024;         // 2048

    const int TB = NN / 128;  // 18

    // ras loss pieces
    k_rowsum<<<NN, 256, 0, stream>>>(grid, rowsum);
    k_colsum<<<(NN + 255) / 256, 256, 0, stream>>>(grid, rowsum, colsum);
    k_ras<<<1024, 256, 0, stream>>>(grid, rowsum, colsum, pRAS);

    // gridloss = sum (G*D) .* (D*G) via WMMA f32
    k_gemm<0><<<dim3(TB, TB), 256, 0, stream>>>(grid, dist, T, nullptr, nullptr);
    k_gemm<1><<<dim3(TB, TB), 256, 0, stream>>>(dist, grid, nullptr, T, pGL);

    // greedy discretization
    k_rowmax<<<NN, 256, 0, stream>>>(grid, rowKey, rowArg);
    k_greedy<<<1, 1024, 0, stream>>>(grid, rowKey, rowArg, perm);
    k_discrete<<<2048, 256, 0, stream>>>(perm, out + 2);
    k_realloss<<<2048, 256, 0, stream>>>(dist, perm, pRL);

    // combine
    k_final<<<1, 256, 0, stream>>>(pGL, TB * TB, pRAS, 1024, pRL, 2048, ratio, out);
}